// DecomposeVelocityFunction_20023137534960
// MI455X (gfx1250) — compile-verified
//
#include <hip/hip_runtime.h>
#include <hip/hip_bf16.h>
#include <math.h>

#define NTOK 65536
#define DD   512
#define LL   8
#define H1D  16
#define H2D  32
#define TT   10
#define NSEG (LL*TT)   /* 80 */

typedef __attribute__((ext_vector_type(2))) float v2f;
typedef __attribute__((ext_vector_type(8))) float v8f;

__device__ __forceinline__ float celu1(float a) {
    return a > 0.f ? a : (__expf(a) - 1.f);
}

__device__ __forceinline__ v8f wmma_f32(v2f a, v2f b, v8f c) {
    // V_WMMA_F32_16X16X4_F32 : D(16x16 f32) = A(16x4 f32) * B(4x16 f32) + C
    return __builtin_amdgcn_wmma_f32_16x16x4_f32(
        /*neg_a=*/false, a, /*neg_b=*/false, b,
        /*c_mod=*/(short)0, c, /*reuse_a=*/false, /*reuse_b=*/false);
}

// ---------------- workspace zeroing ----------------
__global__ void k_zero(int* counts, int* cursor, float* mgsum,
                       float* lin_dot, float* lin_err) {
    int i = blockIdx.x * blockDim.x + threadIdx.x;
    if (i < NSEG) { counts[i] = 0; cursor[i] = 0; }
    if (i < LL)   { lin_dot[i] = 0.f; lin_err[i] = 0.f; }
    for (int j = i; j < NSEG * DD; j += gridDim.x * blockDim.x) mgsum[j] = 0.f;
}

// ---------------- segment histogram ----------------
__global__ void k_count(const int* __restrict__ idx, const int* __restrict__ t,
                        int* counts) {
    int n = blockIdx.x * blockDim.x + threadIdx.x;
    if (n < NTOK) atomicAdd(&counts[idx[n] * TT + t[n]], 1);
}

// ---------------- prefix sums (80 entries, trivial) ----------------
__global__ void k_scan(const int* __restrict__ counts, int* segStart, int* tilesBefore) {
    if (threadIdx.x == 0 && blockIdx.x == 0) {
        int s0 = 0, t0 = 0;
        for (int s = 0; s < NSEG; s++) {
            segStart[s] = s0;
            tilesBefore[s] = t0;
            s0 += counts[s];
            t0 += (counts[s] + 15) / 16;
        }
        segStart[NSEG] = s0;
        tilesBefore[NSEG] = t0;
    }
}

// ---------------- scatter token ids into segment-sorted order ----------------
__global__ void k_scatter(const int* __restrict__ idx, const int* __restrict__ t,
                          const int* __restrict__ segStart, int* cursor, int* order) {
    int n = blockIdx.x * blockDim.x + threadIdx.x;
    if (n < NTOK) {
        int s = idx[n] * TT + t[n];
        int p = segStart[s] + atomicAdd(&cursor[s], 1);
        order[p] = n;
    }
}

// ---------------- main fused tile kernel: 1 wave == 16 tokens of one segment --
__global__ __launch_bounds__(32) void k_tiles(
    const float* __restrict__ v,   const float* __restrict__ x,
    const float* __restrict__ W1,  const float* __restrict__ b1,
    const float* __restrict__ W2,  const float* __restrict__ b2,
    const float* __restrict__ W3,  const float* __restrict__ b3,
    const float* __restrict__ Wg1, const float* __restrict__ bg1,
    const float* __restrict__ Wg2, const float* __restrict__ bg2,
    const float* __restrict__ Wg3, const float* __restrict__ bg3,
    const int* __restrict__ order, const int* __restrict__ segStart,
    const int* __restrict__ tilesBefore,
    float* __restrict__ mgsum, float* __restrict__ lin_dot, float* __restrict__ lin_err)
{
    __shared__ __align__(16) float sh[1536];
    float* h1g = sh;          // 16x16
    float* h1l = sh + 256;    // 16x16
    float* h2g = sh + 512;    // 16x32
    float* h2l = sh + 1024;   // 16x32

    int tile = blockIdx.x;
    int s = 0;
    while (s < NSEG && tilesBefore[s + 1] <= tile) ++s;
    if (s >= NSEG) return;                      // wave-uniform exit
    int lt      = tile - tilesBefore[s];
    int rowBase = segStart[s] + lt * 16;
    int segEnd  = segStart[s + 1];
    int lineage = s / TT;

    int lane = threadIdx.x;
    int half = lane >> 4;      // 0 or 1
    int lm   = lane & 15;      // 0..15

    // token whose row this lane supplies for the A matrix (row = lm)
    int ra   = rowBase + lm;
    int tokA = order[ra < segEnd ? ra : rowBase];
    const float* xrow = x + (size_t)tokA * DD;

    const float* W1l = W1 + (size_t)lineage * DD * H1D;
    const float* b1l = b1 + lineage * H1D;
    const float* W2l = W2 + (size_t)lineage * H1D * H2D;
    const float* b2l = b2 + lineage * H2D;
    const float* W3l = W3 + (size_t)lineage * H2D * DD;
    const float* b3l = b3 + lineage * DD;

    const v8f zero = {0.f, 0.f, 0.f, 0.f, 0.f, 0.f, 0.f, 0.f};

    // ---------- layer 1: x[16x512] @ W[512x16]  (both MLPs) ----------
    v8f cg = zero, cl = zero;
    for (int k0 = 0; k0 < DD; k0 += 4) {
        v2f a = *(const v2f*)(xrow + k0 + 2 * half);
        v2f bg; bg.x = Wg1[(k0 + half) * H1D + lm];   bg.y = Wg1[(k0 + 2 + half) * H1D + lm];
        v2f bl; bl.x = W1l[(k0 + half) * H1D + lm];   bl.y = W1l[(k0 + 2 + half) * H1D + lm];
        cg = wmma_f32(a, bg, cg);
        cl = wmma_f32(a, bl, cl);
    }
    {
        float biasg = bg1[lm], biasl = b1l[lm];
        #pragma unroll
        for (int j = 0; j < 8; j++) {
            h1g[(j + 8 * half) * H1D + lm] = celu1(cg[j] + biasg);
            h1l[(j + 8 * half) * H1D + lm] = celu1(cl[j] + biasl);
        }
    }
    __syncthreads();

    // ---------- layer 2: h1[16x16] @ W2[16x32] ----------
    v8f c20g = zero, c21g = zero, c20l = zero, c21l = zero;
    #pragma unroll
    for (int k0 = 0; k0 < H1D; k0 += 4) {
        v2f ag = *(const v2f*)(&h1g[lm * H1D + k0 + 2 * half]);
        v2f al = *(const v2f*)(&h1l[lm * H1D + k0 + 2 * half]);
        v2f bg0; bg0.x = Wg2[(k0 + half) * H2D + lm];      bg0.y = Wg2[(k0 + 2 + half) * H2D + lm];
        v2f bg1v; bg1v.x = Wg2[(k0 + half) * H2D + 16 + lm]; bg1v.y = Wg2[(k0 + 2 + half) * H2D + 16 + lm];
        v2f bl0; bl0.x = W2l[(k0 + half) * H2D + lm];      bl0.y = W2l[(k0 + 2 + half) * H2D + lm];
        v2f bl1; bl1.x = W2l[(k0 + half) * H2D + 16 + lm]; bl1.y = W2l[(k0 + 2 + half) * H2D + 16 + lm];
        c20g = wmma_f32(ag, bg0, c20g);
        c21g = wmma_f32(ag, bg1v, c21g);
        c20l = wmma_f32(al, bl0, c20l);
        c21l = wmma_f32(al, bl1, c21l);
    }
    __syncthreads();
    {
        #pragma unroll
        for (int j = 0; j < 8; j++) {
            int r = j + 8 * half;
            h2g[r * H2D + lm]      = celu1(c20g[j] + bg2[lm]);
            h2g[r * H2D + 16 + lm] = celu1(c21g[j] + bg2[16 + lm]);
            h2l[r * H2D + lm]      = celu1(c20l[j] + b2l[lm]);
            h2l[r * H2D + 16 + lm] = celu1(c21l[j] + b2l[16 + lm]);
        }
    }
    __syncthreads();

    // rows this lane owns in the C/D layout: M = j + 8*half, col = lm (+d0)
    int   tokR[8];
    float validR[8];
    #pragma unroll
    for (int j = 0; j < 8; j++) {
        int r = rowBase + j + 8 * half;
        bool ok = r < segEnd;
        tokR[j]   = order[ok ? r : rowBase];
        validR[j] = ok ? 1.f : 0.f;
    }

    float acc_dot[8] = {0.f, 0.f, 0.f, 0.f, 0.f, 0.f, 0.f, 0.f};
    float acc_err = 0.f;

    // ---------- layer 3 (16-col chunks) fused with loss epilogue ----------
    for (int d0 = 0; d0 < DD; d0 += 16) {
        v8f dg = zero, dl = zero;
        #pragma unroll
        for (int k0 = 0; k0 < H2D; k0 += 4) {
            v2f ag = *(const v2f*)(&h2g[lm * H2D + k0 + 2 * half]);
            v2f al = *(const v2f*)(&h2l[lm * H2D + k0 + 2 * half]);
            v2f bg; bg.x = Wg3[(k0 + half) * DD + d0 + lm];  bg.y = Wg3[(k0 + 2 + half) * DD + d0 + lm];
            v2f bl; bl.x = W3l[(k0 + half) * DD + d0 + lm];  bl.y = W3l[(k0 + 2 + half) * DD + d0 + lm];
            dg = wmma_f32(ag, bg, dg);
            dl = wmma_f32(al, bl, dl);
        }
        float biasg = bg3[d0 + lm];
        float biasl = b3l[d0 + lm];
        float vgcol = 0.f;
        #pragma unroll
        for (int j = 0; j < 8; j++) {
            float vg   = dg[j] + biasg;
            float vl   = dl[j] + biasl;
            float vv   = v[(size_t)tokR[j] * DD + d0 + lm];
            float diff = vv - vg - vl;
            acc_err    += validR[j] * diff * diff;   // same lineage for whole tile
            acc_dot[j] += vg * vl;                   // per-token partial dot
            vgcol      += validR[j] * vg;            // column partial of segment sum
        }
        vgcol += __shfl_xor(vgcol, 16);              // fold the two row-halves
        if (half == 0) atomicAdd(&mgsum[s * DD + d0 + lm], vgcol);
    }

    // per-token dot across the 16 column-lanes of each half
    float dot2 = 0.f;
    #pragma unroll
    for (int j = 0; j < 8; j++) {
        float dj = acc_dot[j];
        dj += __shfl_xor(dj, 1);
        dj += __shfl_xor(dj, 2);
        dj += __shfl_xor(dj, 4);
        dj += __shfl_xor(dj, 8);
        dot2 += validR[j] * dj * dj;
    }
    dot2 += __shfl_xor(dot2, 16);

    acc_err += __shfl_xor(acc_err, 1);
    acc_err += __shfl_xor(acc_err, 2);
    acc_err += __shfl_xor(acc_err, 4);
    acc_err += __shfl_xor(acc_err, 8);
    acc_err += __shfl_xor(acc_err, 16);

    if (lane == 0) {
        atomicAdd(&lin_dot[lineage], dot2);
        atomicAdd(&lin_err[lineage], acc_err);
    }
}

// ---------------- final reduction: 3 scalar losses ----------------
__global__ void k_final(const int* __restrict__ counts, const float* __restrict__ mgsum,
                        const float* __restrict__ lin_dot, const float* __restrict__ lin_err,
                        float* __restrict__ out)
{
    __shared__ float sdist;
    int tid = threadIdx.x;
    if (tid == 0) sdist = 0.f;
    __syncthreads();
    // 10 timesteps x 28 unordered lineage pairs
    for (int task = tid; task < TT * 28; task += blockDim.x) {
        int tt = task / 28;
        int p  = task % 28;
        int i = 0, j = 1, c = 0;
        for (int a = 0; a < LL; a++)
            for (int b = a + 1; b < LL; b++) {
                if (c == p) { i = a; j = b; }
                c++;
            }
        int cci = counts[i * TT + tt];
        int ccj = counts[j * TT + tt];
        float ci = cci > 0 ? (float)cci : 1.f;
        float cj = ccj > 0 ? (float)ccj : 1.f;
        const float* Mi = mgsum + (size_t)(i * TT + tt) * DD;
        const float* Mj = mgsum + (size_t)(j * TT + tt) * DD;
        float sq = 0.f;
        for (int d = 0; d < DD; d++) {
            float diff = Mi[d] / ci - Mj[d] / cj;
            sq += diff * diff;
        }
        float dist = sq > 0.f ? sqrtf(sq) : 0.f;
        atomicAdd(&sdist, dist);
    }
    __syncthreads();
    if (tid == 0) {
        float recon = 0.f, orth = 0.f;
        for (int l = 0; l < LL; l++) {
            float cl = 0.f;
            for (int t = 0; t < TT; t++) cl += (float)counts[l * TT + t];
            cl = fmaxf(cl, 1.f);
            recon += lin_err[l] / (cl * (float)DD);
            orth  += lin_dot[l] / cl;
        }
        float sim = 2.f * sdist / (float)(LL * (LL - 1)) / (float)TT;
        out[0] = recon;
        out[1] = orth;
        out[2] = sim;
    }
}

extern "C" void kernel_launch(void* const* d_in, const int* in_sizes, int n_in,
                              void* d_out, int out_size, void* d_ws, size_t ws_size,
                              hipStream_t stream)
{
    const float* v   = (const float*)d_in[0];
    const float* x   = (const float*)d_in[1];
    const int*   idx = (const int*)d_in[2];
    const int*   t   = (const int*)d_in[3];
    const float* W1  = (const float*)d_in[4];
    const float* b1  = (const float*)d_in[5];
    const float* W2  = (const float*)d_in[6];
    const float* b2  = (const float*)d_in[7];
    const float* W3  = (const float*)d_in[8];
    const float* b3  = (const float*)d_in[9];
    const float* Wg1 = (const float*)d_in[10];
    const float* bg1 = (const float*)d_in[11];
    const float* Wg2 = (const float*)d_in[12];
    const float* bg2 = (const float*)d_in[13];
    const float* Wg3 = (const float*)d_in[14];
    const float* bg3 = (const float*)d_in[15];

    int*   counts      = (int*)d_ws;
    int*   cursor      = counts + NSEG;
    int*   segStart    = cursor + NSEG;
    int*   tilesBefore = segStart + (NSEG + 1);
    int*   order       = tilesBefore + (NSEG + 1);
    float* mgsum       = (float*)(order + NTOK);
    float* lin_dot     = mgsum + NSEG * DD;
    float* lin_err     = lin_dot + LL;

    k_zero<<<160, 256, 0, stream>>>(counts, cursor, mgsum, lin_dot, lin_err);
    k_count<<<NTOK / 256, 256, 0, stream>>>(idx, t, counts);
    k_scan<<<1, 1, 0, stream>>>(counts, segStart, tilesBefore);
    k_scatter<<<NTOK / 256, 256, 0, stream>>>(idx, t, segStart, cursor, order);

    int maxTiles = NTOK / 16 + NSEG;   // upper bound incl. per-segment ragged tiles
    k_tiles<<<maxTiles, 32, 0, stream>>>(v, x, W1, b1, W2, b2, W3, b3,
                                         Wg1, bg1, Wg2, bg2, Wg3, bg3,
                                         order, segStart, tilesBefore,
                                         mgsum, lin_dot, lin_err);

    k_final<<<1, 256, 0, stream>>>(counts, mgsum, lin_dot, lin_err, (float*)d_out);
}